// SelfModifyingTitans_14035953124011
// MI455X (gfx1250) — compile-verified
//
#include <hip/hip_runtime.h>
#include <hip/hip_bf16.h>

// ---------------------------------------------------------------------------
// SelfModifyingTitans on MI455X (gfx1250, wave32)
//   q/k/v = x @ W{q,k,v} + b   -> f32 WMMA GEMM, async-to-LDS double buffered
//   delta-rule scan over S     -> 1 WG per batch, W in VGPRs, shfl butterfly,
//                                 1 barrier/step, async-staged q/k/v
//   out   = ys @ Wo + bo       -> same WMMA GEMM
// ---------------------------------------------------------------------------

typedef __attribute__((ext_vector_type(2))) float v2f;
typedef __attribute__((ext_vector_type(8))) float v8f;

#define BB   8
#define SS   2048
#define DIMD 2048
#define HDH  128
#define MM   (BB * SS)        // 16384 rows

#define TM   64               // workgroup M tile
#define TN   64               // workgroup N tile (2 x 32-col slabs)
#define KC   16               // K chunk staged in LDS
#define APAD 20               // A row stride in dwords (16B aligned; halves hit
                              // banks ==0,1 (mod 4) vs ==2,3 -> conflict-free)
#define BROW 160              // B pair-row stride in dwords (160%64==32 ->
                              // lane halves cover disjoint 32-bank sets)

// Async global->LDS copies (CDNA5 path, ASYNCcnt-tracked).
// Generic->LDS offset: the LDS aperture maps addr[31:0] to the LDS byte offset.
__device__ __forceinline__ void async_ld_b128(const void* lds, const float* g) {
  unsigned l = (unsigned)(unsigned long long)lds;
  asm volatile("global_load_async_to_lds_b128 %0, %1, off"
               :: "v"(l), "v"(g) : "memory");
}
__device__ __forceinline__ void async_ld_b32(const void* lds, const float* g) {
  unsigned l = (unsigned)(unsigned long long)lds;
  asm volatile("global_load_async_to_lds_b32 %0, %1, off"
               :: "v"(l), "v"(g) : "memory");
}
__device__ __forceinline__ void wait_async0() {
  asm volatile("s_wait_asynccnt 0" ::: "memory");
}

// C[M,N] = A[M,K] * B[K,N] + bias[N]   (lda=K, ldb=ldc=N; all tile-divisible)
__global__ __launch_bounds__(256)
void titans_gemm_wmma_f32(const float* __restrict__ A,
                          const float* __restrict__ B,
                          const float* __restrict__ bias,
                          float* __restrict__ C,
                          int M, int N, int K) {
  __shared__ float As[2][TM][APAD];          // row-major A tile
  __shared__ float Bs[2][KC / 2][BROW];      // (k,n) at [k>>1][2n + (k&1)]

  const int t    = threadIdx.x;
  const int lane = t & 31;
  const int wave = t >> 5;          // 0..7
  const int wm   = wave >> 1;       // 0..3  (16-row sub-tile)
  const int wn   = wave & 1;        // 0..1  (32-col slab)
  const int m0   = blockIdx.x * TM;
  const int n0   = blockIdx.y * TN;

  const int lm  = lane & 15;        // row/col within 16
  const int lk2 = (lane >> 4) * 2;  // K pair select per lane half (even)

  // staging coordinates
  const int ar = t >> 2, ac = (t & 3) * 4;    // A: 64 x 16, float4/thread
  const int bk = t >> 4, bn = (t & 15) * 4;   // B: 16 x 64, 4 floats/thread

  // running global pointers (strength-reduced; bumped per issued chunk)
  const float* gAp = A + (size_t)(m0 + ar) * K + ac;
  const float* gBp = B + (size_t)bk * N + n0 + bn;

  v8f acc0 = {0.f,0.f,0.f,0.f,0.f,0.f,0.f,0.f};
  v8f acc1 = {0.f,0.f,0.f,0.f,0.f,0.f,0.f,0.f};

  auto issue_chunk = [&](int sel) {
    async_ld_b128(&As[sel][ar][ac], gAp);
#pragma unroll
    for (int j = 0; j < 4; ++j)    // pair-interleaved scatter: stride-2 LDS
      async_ld_b32(&Bs[sel][bk >> 1][(bn + j) * 2 + (bk & 1)], gBp + j);
    gAp += KC;
    gBp += (size_t)KC * N;
  };

  issue_chunk(0);
  int sel = 0;
  for (int kc = 0; kc < K; kc += KC, sel ^= 1) {
    wait_async0();            // chunk `sel` landed (own wave's async ops)
    __syncthreads();          // ... and everyone else's too
    if (kc + KC < K) issue_chunk(sel ^ 1);   // prefetch next chunk

#pragma unroll
    for (int kk = 0; kk < KC; kk += 4) {
      const int kp = (kk + lk2) >> 1;        // K pair index (kk+lk2 is even)
      // A 16x4 frag: lane holds row lm, K = kk+lk2, +1  -> one b64 load
      v2f a  = *(const v2f*)&As[sel][wm * 16 + lm][kk + lk2];
      // B 4x16 frags: pair-interleaved rows -> direct b64 into operand pair
      v2f b0 = *(const v2f*)&Bs[sel][kp][(wn * 32 + lm) * 2];
      v2f b1 = *(const v2f*)&Bs[sel][kp][(wn * 32 + 16 + lm) * 2];
      acc0 = __builtin_amdgcn_wmma_f32_16x16x4_f32(
          false, a, false, b0, (short)0, acc0, false, false);
      acc1 = __builtin_amdgcn_wmma_f32_16x16x4_f32(
          false, a, false, b1, (short)0, acc1, false, false);
    }
    // no trailing barrier: next async targets the other buffer; buffer reuse
    // at chunk+2 is ordered behind barrier(chunk+1), which each wave reaches
    // only after its DS reads here were consumed by the WMMAs.
  }

  // C/D 16x16 layout: acc[r] -> row (r + 8*(lane>>4)), col (lane&15)
  const int col0  = n0 + wn * 32 + lm;
  const int rbase = m0 + wm * 16 + (lane >> 4) * 8;
  const float bv0 = bias[col0];
  const float bv1 = bias[col0 + 16];
#pragma unroll
  for (int r = 0; r < 8; ++r) {
    C[(size_t)(rbase + r) * N + col0]      = acc0[r] + bv0;
    C[(size_t)(rbase + r) * N + col0 + 16] = acc1[r] + bv1;
  }
}

// ---------------------------------------------------------------------------
// Delta-rule scan: one workgroup per batch, 1024 threads (32 waves).
// thread t: row i = t>>3, column group g = t&7 (16 cols), W slice in VGPRs.
// Per step: y = W q, err = W k - v, W -= lr * err k^T.
// Row partials live in one wave -> shfl_xor butterfly, no reduction barrier.
// ---------------------------------------------------------------------------
__global__ __launch_bounds__(1024)
void titans_scan(const float* __restrict__ q, const float* __restrict__ k,
                 const float* __restrict__ v, const float* __restrict__ state,
                 const float* __restrict__ lr_p, float* __restrict__ ys,
                 float* __restrict__ Wout) {
  __shared__ float stg[2][3 * HDH];   // [buf][ q(128) | k(128) | v(128) ]

  const int b  = blockIdx.x;
  const int t  = threadIdx.x;
  const int g  = t & 7;
  const int i  = t >> 3;
  const int j0 = g * 16;
  const float lr = lr_p[0];

  float W[16];
#pragma unroll
  for (int c = 0; c < 16; ++c)
    W[c] = state[(size_t)b * (HDH * HDH) + i * HDH + j0 + c];

  const float* qb = q + (size_t)b * SS * HDH;
  const float* kb = k + (size_t)b * SS * HDH;
  const float* vb = v + (size_t)b * SS * HDH;
  float*       yb = ys + (size_t)b * SS * HDH;

  // threads 0..383 each own one staged element per step
  const float* src = nullptr;
  if (t < 128)      src = qb + t;
  else if (t < 256) src = kb + (t - 128);
  else if (t < 384) src = vb + (t - 256);

  if (t < 384) async_ld_b32(&stg[0][t], src);        // stage step 0

  for (int s = 0; s < SS; ++s) {
    const int sel = s & 1;
    wait_async0();
    __syncthreads();
    if (t < 384 && s + 1 < SS)                       // prefetch step s+1
      async_ld_b32(&stg[sel ^ 1][t], src + (size_t)(s + 1) * HDH);

    const float* qs = stg[sel];
    const float* ks = stg[sel] + HDH;
    const float* vs = stg[sel] + 2 * HDH;

    float sq = 0.f, sk = 0.f;
#pragma unroll
    for (int c = 0; c < 16; ++c) {
      const float w = W[c];
      sq = fmaf(w, qs[j0 + c], sq);
      sk = fmaf(w, ks[j0 + c], sk);
    }
    // butterfly across the 8 column groups (lane bits 0..2 of this wave)
#pragma unroll
    for (int m = 1; m < 8; m <<= 1) {
      sq += __shfl_xor(sq, m, 32);
      sk += __shfl_xor(sk, m, 32);
    }
    const float e = lr * (sk - vs[i]);               // all 8 lanes agree
    if (g == 0) yb[(size_t)s * HDH + i] = sq;

#pragma unroll
    for (int c = 0; c < 16; ++c)
      W[c] = fmaf(-e, ks[j0 + c], W[c]);
    // single barrier per step: buffer `sel` is only rewritten at step s+2,
    // behind barrier(s+1) which orders after this step's consumed LDS reads.
  }

#pragma unroll
  for (int c = 0; c < 16; ++c)
    Wout[(size_t)b * (HDH * HDH) + i * HDH + j0 + c] = W[c];
}

// ---------------------------------------------------------------------------
// inputs: 0:x 1:state 2:Wq 3:bq 4:Wk 5:bk 6:Wv 7:bv 8:Wo 9:bo 10:lr
// d_out: out [B,S,DIM] ++ W_final [B,HD,HD]
// ---------------------------------------------------------------------------
extern "C" void kernel_launch(void* const* d_in, const int* in_sizes, int n_in,
                              void* d_out, int out_size, void* d_ws, size_t ws_size,
                              hipStream_t stream) {
  const float* x     = (const float*)d_in[0];
  const float* state = (const float*)d_in[1];
  const float* Wq    = (const float*)d_in[2];
  const float* bq    = (const float*)d_in[3];
  const float* Wk    = (const float*)d_in[4];
  const float* bk    = (const float*)d_in[5];
  const float* Wv    = (const float*)d_in[6];
  const float* bv    = (const float*)d_in[7];
  const float* Wo    = (const float*)d_in[8];
  const float* bo    = (const float*)d_in[9];
  const float* lr    = (const float*)d_in[10];

  float* out    = (float*)d_out;
  float* Wfinal = out + (size_t)BB * SS * DIMD;

  const size_t BSH = (size_t)BB * SS * HDH;   // 2,097,152 floats
  float* q  = (float*)d_ws;
  float* k  = q + BSH;
  float* v  = k + BSH;
  float* ys = v + BSH;                        // 32 MB of workspace total

  dim3 blk(256);
  dim3 gproj(MM / TM, HDH / TN);              // 256 x 2
  titans_gemm_wmma_f32<<<gproj, blk, 0, stream>>>(x, Wq, bq, q, MM, HDH, DIMD);
  titans_gemm_wmma_f32<<<gproj, blk, 0, stream>>>(x, Wk, bk, k, MM, HDH, DIMD);
  titans_gemm_wmma_f32<<<gproj, blk, 0, stream>>>(x, Wv, bv, v, MM, HDH, DIMD);

  titans_scan<<<BB, 1024, 0, stream>>>(q, k, v, state, lr, ys, Wfinal);

  dim3 gout(MM / TM, DIMD / TN);              // 256 x 32
  titans_gemm_wmma_f32<<<gout, blk, 0, stream>>>(ys, Wo, bo, out, MM, DIMD, HDH);
}